// DepthSeparableConv2d_conv2_1_37572373905412
// MI455X (gfx1250) — compile-verified
//
#include <hip/hip_runtime.h>
#include <hip/hip_bf16.h>

typedef __attribute__((ext_vector_type(2))) float v2f;
typedef __attribute__((ext_vector_type(8))) float v8f;

#define HW_   12544           // 112*112
#define WIDTH 112
#define CIN   64
#define COUT  128
#define NB    32
#define EPS_  1e-5f

// ---------------------------------------------------------------- k_zero ---
__global__ __launch_bounds__(256) void k_zero(float* __restrict__ p, int n) {
    int i = blockIdx.x * 256 + threadIdx.x;
    if (i < n) p[i] = 0.0f;
}

// -------------------------------------------------------------- k_dwconv ---
// One block per (b,c) plane. Input plane staged in LDS; fused
// conv3x3 + bias + BN1 + ReLU; block max -> plane cut at 4.0; write y.
__global__ __launch_bounds__(256) void k_dwconv(
    const float* __restrict__ x,  const float* __restrict__ dw_w,
    const float* __restrict__ dw_b,
    const float* __restrict__ g,  const float* __restrict__ bb,
    const float* __restrict__ mm, const float* __restrict__ vv,
    float* __restrict__ y)
{
    __shared__ float tile[HW_];
    __shared__ float red[256];
    const int plane = blockIdx.x;      // b*64 + c
    const int c     = plane & (CIN - 1);
    const int tid   = threadIdx.x;

    const float* xp = x + (size_t)plane * HW_;
    for (int i = tid; i < HW_; i += 256) tile[i] = xp[i];

    float wreg[9];
#pragma unroll
    for (int j = 0; j < 9; ++j) wreg[j] = dw_w[c * 9 + j];
    const float inv1 = g[c] * rsqrtf(vv[c] + EPS_);
    const float add1 = bb[c] - mm[c] * inv1 + dw_b[c] * inv1;

    __syncthreads();

    float out[49];
    float tmax = 0.0f;
#pragma unroll
    for (int j = 0; j < 49; ++j) {
        const int i = tid + j * 256;
        const int h = i / WIDTH;
        const int w = i - h * WIDTH;
        float s = 0.0f;
#pragma unroll
        for (int kh = 0; kh < 3; ++kh) {
            const int hh = h + kh - 1;
            const bool hok = (hh >= 0) && (hh < WIDTH);
#pragma unroll
            for (int kw = 0; kw < 3; ++kw) {
                const int ww = w + kw - 1;
                const bool ok = hok && (ww >= 0) && (ww < WIDTH);
                const float xv = ok ? tile[hh * WIDTH + ww] : 0.0f;
                s = fmaf(xv, wreg[kh * 3 + kw], s);
            }
        }
        const float val = fmaxf(fmaf(s, inv1, add1), 0.0f);
        out[j] = val;
        tmax = fmaxf(tmax, val);
    }

    red[tid] = tmax;
    __syncthreads();
    for (int s = 128; s > 0; s >>= 1) {
        if (tid < s) red[tid] = fmaxf(red[tid], red[tid + s]);
        __syncthreads();
    }
    const float keep = (red[0] < 4.0f) ? 0.0f : 1.0f;

    float* yp = y + (size_t)plane * HW_;
#pragma unroll
    for (int j = 0; j < 49; ++j) yp[tid + j * 256] = out[j] * keep;
}

// -------------------------------------------------------------- k_pwconv ---
// Pointwise 1x1 conv as WMMA GEMM: Z[o,n] = sum_c W[o,c] * Y[c,n], per batch.
// Block: 256 thr = 8 waves; wave w owns output-channel tile [16w,16w+16),
// block covers 64 spatial positions. Y tile staged in LDS (row stride 72 so
// the two 16-lane halves of each B load hit disjoint bank sets).
#define YSTR 72
__global__ __launch_bounds__(256) void k_pwconv(
    const float* __restrict__ y,  const float* __restrict__ pw_w,
    const float* __restrict__ pw_b,
    const float* __restrict__ g,  const float* __restrict__ bb,
    const float* __restrict__ mm, const float* __restrict__ vv,
    float* __restrict__ z, float* __restrict__ maxbuf)
{
    __shared__ float ytile[CIN * YSTR];
    const int b     = blockIdx.y;
    const int nbase = blockIdx.x * 64;
    const int tid   = threadIdx.x;
    const int wave  = tid >> 5;
    const int lane  = tid & 31;

    // cooperative load: 64 channels x 64 spatial
    for (int i = tid; i < CIN * 64; i += 256) {
        const int cc = i >> 6;
        const int nn = i & 63;
        ytile[cc * YSTR + nn] = y[((size_t)b * CIN + cc) * HW_ + nbase + nn];
    }

    // A-matrix (weights) per ISA 16x4 f32 layout:
    //   lanes 0-15: M=lane, K = {0,1}; lanes 16-31: M=lane-16, K = {2,3}
    const int m0   = wave * 16;
    const int row  = m0 + (lane & 15);
    const int koff = (lane >> 4) << 1;                // 0 or 2
    v2f a[16];
#pragma unroll
    for (int kc = 0; kc < 16; ++kc) {
        a[kc].x = pw_w[row * CIN + kc * 4 + koff];
        a[kc].y = pw_w[row * CIN + kc * 4 + koff + 1];
    }

    // BN2 coeffs for the 8 channels this lane holds in D
    // (D vgpr r, lanes 0-15 -> M=r, lanes 16-31 -> M=r+8)
    const int half8 = (lane >> 4) << 3;
    float inv2[8], add2[8];
#pragma unroll
    for (int r = 0; r < 8; ++r) {
        const int o = m0 + r + half8;
        const float iv = g[o] * rsqrtf(vv[o] + EPS_);
        inv2[r] = iv;
        add2[r] = bb[o] - mm[o] * iv + pw_b[o] * iv;
    }

    __syncthreads();

    const int ncol = lane & 15;
    for (int nt = 0; nt < 4; ++nt) {
        v8f acc = {};
#pragma unroll
        for (int kc = 0; kc < 16; ++kc) {
            // B-matrix 4x16 f32: lanes 0-15: K={0,1}, N=lane;
            //                    lanes 16-31: K={2,3}, N=lane-16
            v2f bv;
            bv.x = ytile[(kc * 4 + koff)     * YSTR + nt * 16 + ncol];
            bv.y = ytile[(kc * 4 + koff + 1) * YSTR + nt * 16 + ncol];
            acc = __builtin_amdgcn_wmma_f32_16x16x4_f32(
                false, a[kc], false, bv, (short)0, acc, false, false);
        }

        // fused bias + BN2 + ReLU epilogue; per-channel max via shfl width 16
        float cmax[8];
#pragma unroll
        for (int r = 0; r < 8; ++r) {
            const int o = m0 + r + half8;
            float val = fmaxf(fmaf(acc[r], inv2[r], add2[r]), 0.0f);
            z[((size_t)b * COUT + o) * HW_ + nbase + nt * 16 + ncol] = val;
            float mx = val;
            mx = fmaxf(mx, __shfl_xor(mx, 1, 16));
            mx = fmaxf(mx, __shfl_xor(mx, 2, 16));
            mx = fmaxf(mx, __shfl_xor(mx, 4, 16));
            mx = fmaxf(mx, __shfl_xor(mx, 8, 16));
            cmax[r] = mx;
        }
        if (ncol == 0) {
#pragma unroll
            for (int r = 0; r < 8; ++r) {
                const int o = m0 + r + half8;
                // values are >= 0 post-ReLU: int compare == float compare
                atomicMax((int*)&maxbuf[b * COUT + o], __float_as_int(cmax[r]));
            }
        }
    }
}

// ----------------------------------------------------------------- k_cut ---
// In-place plane cut on z (threshold 0.001), vectorized float4.
__global__ __launch_bounds__(256) void k_cut(float* __restrict__ z,
                                             const float* __restrict__ maxbuf)
{
    const size_t base = ((size_t)blockIdx.x * 256 + threadIdx.x) * 4;
    if (base >= (size_t)NB * COUT * HW_) return;
    const int plane = (int)(base / HW_);          // HW_ % 4 == 0
    const float keep = (maxbuf[plane] < 0.001f) ? 0.0f : 1.0f;
    float4 v = *(float4*)(z + base);
    v.x *= keep; v.y *= keep; v.z *= keep; v.w *= keep;
    *(float4*)(z + base) = v;
}

// ---------------------------------------------------------------- launch ---
extern "C" void kernel_launch(void* const* d_in, const int* in_sizes, int n_in,
                              void* d_out, int out_size, void* d_ws, size_t ws_size,
                              hipStream_t stream) {
    const float* x     = (const float*)d_in[0];
    const float* dw_w  = (const float*)d_in[1];
    const float* dw_b  = (const float*)d_in[2];
    const float* bn1_g = (const float*)d_in[3];
    const float* bn1_b = (const float*)d_in[4];
    const float* bn1_m = (const float*)d_in[5];
    const float* bn1_v = (const float*)d_in[6];
    const float* pw_w  = (const float*)d_in[7];
    const float* pw_b  = (const float*)d_in[8];
    const float* bn2_g = (const float*)d_in[9];
    const float* bn2_b = (const float*)d_in[10];
    const float* bn2_m = (const float*)d_in[11];
    const float* bn2_v = (const float*)d_in[12];
    float* out = (float*)d_out;

    float* y      = (float*)d_ws;                        // NB*CIN*HW_ floats
    float* maxbuf = y + (size_t)NB * CIN * HW_;          // NB*COUT floats

    k_zero<<<(NB * COUT + 255) / 256, 256, 0, stream>>>(maxbuf, NB * COUT);
    k_dwconv<<<NB * CIN, 256, 0, stream>>>(x, dw_w, dw_b,
                                           bn1_g, bn1_b, bn1_m, bn1_v, y);
    k_pwconv<<<dim3(HW_ / 64, NB), 256, 0, stream>>>(y, pw_w, pw_b,
                                                     bn2_g, bn2_b, bn2_m, bn2_v,
                                                     out, maxbuf);
    const size_t nvec4 = ((size_t)NB * COUT * HW_) / 4;
    k_cut<<<(unsigned)((nvec4 + 255) / 256), 256, 0, stream>>>(out, maxbuf);
}